// SingleStageDetector_64742337020718
// MI455X (gfx1250) — compile-verified
//
#include <hip/hip_runtime.h>

// Problem constants from the reference
#define BB 8
#define AA 9
#define HH 56
#define WW 56
#define NN 64
#define HWC (HH * WW)        // 3136
#define PP  (AA * HWC)       // 28224 proposals per batch (divisible by 16)

typedef __attribute__((ext_vector_type(2))) float v2f;
typedef __attribute__((ext_vector_type(8))) float v8f;

// One wave (32 lanes) owns a tile of 16 proposals x all 64 bboxes.
// Block = 256 threads = 8 waves. Grid = B*(P/16)/8 = 1764 blocks exactly.
__global__ __launch_bounds__(256)
void ssd_proposal_iou_kernel(const float* __restrict__ anc,      // (A,2)
                             const float* __restrict__ grid,     // (B,H,W,2)
                             const float* __restrict__ offsets,  // (B,A,H,W,4)
                             const float* __restrict__ bboxes,   // (B,N,5)
                             float* __restrict__ out_prop,       // (B,P,4)
                             float* __restrict__ out_iou)        // (B,P,N)
{
    const int lane = threadIdx.x & 31;
    const int wave = threadIdx.x >> 5;
    const int tile = blockIdx.x * 8 + wave;   // global 16-row tile id
    const int b    = tile / (PP / 16);
    const int t    = tile - b * (PP / 16);
    const int row0 = t * 16;

    // ---- proposal for row m = lane & 15 (both lane-halves compute the same
    //      16 rows; needed for the WMMA A-operand and as shfl sources) ----
    const int m = lane & 15;
    const int p = row0 + m;                // flat (a,h,w) index
    const int a = p / HWC;
    const int r = p - a * HWC;             // h*W + w

    const float gx  = grid[((b * HWC + r) << 1) + 0];
    const float gy  = grid[((b * HWC + r) << 1) + 1];
    const float hw_ = anc[2 * a + 0] * 0.5f;
    const float hh_ = anc[2 * a + 1] * 0.5f;

    const float* off = offsets + (((size_t)b * PP + p) << 2);
    const float o0 = off[0], o1 = off[1], o2 = off[2], o3 = off[3];

    // anchors: [gx-hw, gy-hh, gx+hw, gy+hh] -> cx=gx+o0, cy=gy+o1,
    // w = 2*hw*exp(o2), h = 2*hh*exp(o3)
    const float cx = gx + o0;
    const float cy = gy + o1;
    const float wd = (2.0f * hw_) * __expf(o2);
    const float ht = (2.0f * hh_) * __expf(o3);

    const float px0 = cx - 0.5f * wd;
    const float py0 = cy - 0.5f * ht;
    const float px1 = cx + 0.5f * wd;
    const float py1 = cy + 0.5f * ht;
    const float pa  = wd * ht;             // proposal area (exact: x1-x0 == wd)

    if (lane < 16) {                        // coalesced float4 proposal store
        float4 pr = make_float4(px0, py0, px1, py1);
        *(float4*)(out_prop + (((size_t)b * PP + p) << 2)) = pr;
    }

    // ---- WMMA A operand: A[m][k] = {pa_m, 1, 0, 0}  (16x4 f32, 2 VGPRs)
    // lanes 0-15 hold K=0,1 ; lanes 16-31 hold K=2,3 (zeros)
    v2f av;
    av.x = (lane < 16) ? pa   : 0.0f;
    av.y = (lane < 16) ? 1.0f : 0.0f;

    const int hi = lane >> 4;               // which 8-row half this lane owns

    // ---- hoisted cross-lane proposal coords: chunk-invariant, do the 32
    //      bpermutes once per tile instead of once per 16-col chunk ----
    float qx0[8], qy0[8], qx1[8], qy1[8];
    #pragma unroll
    for (int v = 0; v < 8; ++v) {
        const int mr = v + 8 * hi;          // row this lane owns in D VGPR v
        qx0[v] = __shfl(px0, mr, 32);
        qy0[v] = __shfl(py0, mr, 32);
        qx1[v] = __shfl(px1, mr, 32);
        qy1[v] = __shfl(py1, mr, 32);
    }

    const float* bbb = bboxes + (size_t)b * NN * 5;

    for (int chunk = 0; chunk < 4; ++chunk) {
        // this lane's bbox column n = chunk*16 + (lane & 15)
        const int n = chunk * 16 + m;
        const float* bbp = bbb + (size_t)n * 5;
        const float bx0 = bbp[0], by0 = bbp[1], bx1 = bbp[2], by1 = bbp[3];
        const float ba  = (bx1 - bx0) * (by1 - by0);

        // ---- WMMA B operand: B[k][n] = {1, ba_n, 0, 0}^T (4x16 f32, 2 VGPRs)
        v2f bv;
        bv.x = (lane < 16) ? 1.0f : 0.0f;   // K=0 row of ones
        bv.y = (lane < 16) ? ba   : 0.0f;   // K=1 row = bbox areas

        v8f c = {};
        // D[m][n] = pa_m + ba_n  (outer-sum via rank-2 matmul on matrix pipe)
        v8f d = __builtin_amdgcn_wmma_f32_16x16x4_f32(
            /*neg_a=*/false, av, /*neg_b=*/false, bv,
            /*c_mod=*/(short)0, c, /*reuse_a=*/false, /*reuse_b=*/false);

        // D layout: lane owns column N = lane&15, rows M = v + 8*hi
        #pragma unroll
        for (int v = 0; v < 8; ++v) {
            const int mr    = v + 8 * hi;
            const float iw    = fminf(qx1[v], bx1) - fmaxf(qx0[v], bx0);
            const float ih    = fminf(qy1[v], by1) - fmaxf(qy0[v], by0);
            const float inter = fmaxf(iw, 0.0f) * fmaxf(ih, 0.0f);
            const float uni   = d[v] - inter;          // (pa+ba) - inter
            // fast reciprocal (v_rcp_f32): ~1 ulp, avoids the IEEE divide
            // expansion (div_scale/div_fmas/div_fixup ~10 VALU ops/pair)
            const float iou   = inter * __builtin_amdgcn_rcpf(uni);

            // coalesced: consecutive lanes -> consecutive n
            out_iou[((size_t)(b * PP + row0 + mr)) * NN + n] = iou;
        }
    }
}

extern "C" void kernel_launch(void* const* d_in, const int* in_sizes, int n_in,
                              void* d_out, int out_size, void* d_ws, size_t ws_size,
                              hipStream_t stream) {
    (void)in_sizes; (void)n_in; (void)out_size; (void)d_ws; (void)ws_size;

    const float* anc     = (const float*)d_in[0];
    const float* grid    = (const float*)d_in[1];
    const float* offsets = (const float*)d_in[2];
    const float* bboxes  = (const float*)d_in[3];

    float* out_prop = (float*)d_out;                       // B*P*4 floats
    float* out_iou  = out_prop + (size_t)BB * PP * 4;      // B*P*N floats

    const int tiles  = BB * (PP / 16);    // 14112
    const int blocks = tiles / 8;         // 1764, exact

    ssd_proposal_iou_kernel<<<blocks, 256, 0, stream>>>(
        anc, grid, offsets, bboxes, out_prop, out_iou);
}